// OnlyLinear_5_14671608283875
// MI455X (gfx1250) — compile-verified
//
#include <hip/hip_runtime.h>
#include <hip/hip_bf16.h>

// fp32 WMMA operand types: 16x16x4 f32 => A,B = 2 VGPRs/lane, C/D = 8 VGPRs/lane
typedef __attribute__((ext_vector_type(2))) float v2f;
typedef __attribute__((ext_vector_type(8))) float v8f;

// Fixed MLP chain: 784->26->12->11->10->10->10->10->10, batch 1, all fp32.
// Single workgroup (8 wave32), fully fused. Layer 0 (26x784 GEMV) runs on the
// matrix pipe via V_WMMA_F32_16X16X4_F32 (full fp32 precision); layers 1..7
// are tiny serial dots (one thread per output neuron) through LDS.
//
// Layer-0 mapping per wave: A(16x4) = W0 tile rows, B(4x16) = x chunk
// broadcast across all 16 N columns, D accumulates the GEMV partials.
// Rows 26..31 alias row 25 (clamped pointer, no masking) — their garbage
// lands in D rows the reduction never reads, so no select/EXEC games needed.
__global__ __launch_bounds__(256, 1)
void mlp_fused_kernel(const float* __restrict__ x,
                      const float* __restrict__ W0, const float* __restrict__ b0,
                      const float* __restrict__ W1, const float* __restrict__ b1,
                      const float* __restrict__ W2, const float* __restrict__ b2,
                      const float* __restrict__ W3, const float* __restrict__ b3,
                      const float* __restrict__ W4, const float* __restrict__ b4,
                      const float* __restrict__ W5, const float* __restrict__ b5,
                      const float* __restrict__ W6, const float* __restrict__ b6,
                      const float* __restrict__ W7, const float* __restrict__ b7,
                      float* __restrict__ out)
{
    __shared__ float partial[8][16];   // per-wave partial GEMV rows for layer 0
    __shared__ float actA[32];         // activation ping
    __shared__ float actB[32];         // activation pong

    const int tid  = threadIdx.x;
    const int wave = tid >> 5;         // 8 waves of 32
    const int lane = tid & 31;

    // Warm the small later-layer weights into cache while the WMMA loop runs.
    if (wave == 0) {
        const float* pf[7] = { W1, W2, W3, W4, W5, W6, W7 };
        if (lane < 7) {
            __builtin_prefetch(pf[lane], 0, 3);
            __builtin_prefetch(pf[lane] + 64, 0, 3);   // next cacheline
        }
    }

    // ---------------- Layer 0: out26 = W0(26x784) @ x + b0, via fp32 WMMA ---
    // 2 M-tiles (rows 0-15 / 16-25) x 4 K-slots of 196 columns (49 k-steps).
    const int mtile = wave >> 2;                 // 0 or 1
    const int kslot = wave & 3;                  // K chunk owner
    const int m     = mtile * 16 + (lane & 15);  // output row this lane feeds
    const int koff  = (lane >> 4) * 2;           // lanes 0-15: K{0,1}; 16-31: K{2,3}
    const int mr    = (m < 26) ? m : 25;         // clamp; aliased rows never read back
    const float* __restrict__ wrow = W0 + mr * 784;
    const int kbase = kslot * 196 + koff;        // even => 8B-aligned float2 loads

    v8f acc = {0.f, 0.f, 0.f, 0.f, 0.f, 0.f, 0.f, 0.f};

    // Software-pipelined accumulation: keep next A/B fragments in flight
    // while the current WMMA executes (avoids s_wait_loadcnt 0 before wmma).
    v2f a = *(const v2f*)(wrow + kbase);         // one global_load_b64 each
    v2f b = *(const v2f*)(x    + kbase);
    #pragma unroll 8
    for (int t = 0; t < 48; ++t) {
        const int kn = kbase + (t + 1) * 4;
        v2f an = *(const v2f*)(wrow + kn);
        v2f bn = *(const v2f*)(x    + kn);
        acc = __builtin_amdgcn_wmma_f32_16x16x4_f32(
                  false, a, false, b, (short)0, acc, false, false);
        a = an; b = bn;
    }
    acc = __builtin_amdgcn_wmma_f32_16x16x4_f32(
              false, a, false, b, (short)0, acc, false, false);

    // Every D column holds the same partial GEMV (B replicated across N).
    // Extract column N=0: lane 0 carries rows 0..7 (VGPR j -> M=j), lane 16
    // carries rows 8..15.
    if ((lane & 15) == 0) {
        const int rbase = (lane >> 4) * 8;
        #pragma unroll
        for (int j = 0; j < 8; ++j) partial[wave][rbase + j] = acc[j];
    }
    __syncthreads();

    // Cross-wave K reduction + bias -> actA[0..25]
    if (tid < 26) {
        const int wbase = (tid < 16) ? 0 : 4;
        const int mm    = (tid < 16) ? tid : tid - 16;
        float s = b0[tid];
        #pragma unroll
        for (int w = 0; w < 4; ++w) s += partial[wbase + w][mm];
        actA[tid] = s;
    }
    __syncthreads();

    // ---------------- Layers 1..7: tiny serial fp32 dots through LDS --------
    const float* Ws[7]   = { W1, W2, W3, W4, W5, W6, W7 };
    const float* bs[7]   = { b1, b2, b3, b4, b5, b6, b7 };
    const int    ins[7]  = { 26, 12, 11, 10, 10, 10, 10 };
    const int    outs[7] = { 12, 11, 10, 10, 10, 10, 10 };

    float* cur = actA;
    float* nxt = actB;
    #pragma unroll
    for (int L = 0; L < 7; ++L) {
        const int fin  = ins[L];
        const int fout = outs[L];
        if (tid < fout) {
            float s = bs[L][tid];
            const float* __restrict__ wr = Ws[L] + tid * fin;
            for (int k = 0; k < fin; ++k) s += wr[k] * cur[k];
            nxt[tid] = s;
        }
        __syncthreads();
        float* tmp = cur; cur = nxt; nxt = tmp;
    }

    if (tid < 10) out[tid] = cur[tid];
}

extern "C" void kernel_launch(void* const* d_in, const int* in_sizes, int n_in,
                              void* d_out, int out_size, void* d_ws, size_t ws_size,
                              hipStream_t stream)
{
    // setup_inputs order: x, W0, b0, W1, b1, ..., W7, b7
    const float* x  = (const float*)d_in[0];
    const float* W0 = (const float*)d_in[1];  const float* b0 = (const float*)d_in[2];
    const float* W1 = (const float*)d_in[3];  const float* b1 = (const float*)d_in[4];
    const float* W2 = (const float*)d_in[5];  const float* b2 = (const float*)d_in[6];
    const float* W3 = (const float*)d_in[7];  const float* b3 = (const float*)d_in[8];
    const float* W4 = (const float*)d_in[9];  const float* b4 = (const float*)d_in[10];
    const float* W5 = (const float*)d_in[11]; const float* b5 = (const float*)d_in[12];
    const float* W6 = (const float*)d_in[13]; const float* b6 = (const float*)d_in[14];
    const float* W7 = (const float*)d_in[15]; const float* b7 = (const float*)d_in[16];
    float* out = (float*)d_out;

    mlp_fused_kernel<<<1, 256, 0, stream>>>(x, W0, b0, W1, b1, W2, b2, W3, b3,
                                            W4, b4, W5, b5, W6, b6, W7, b7, out);
}